// MenuLoss_2929167696249
// MI455X (gfx1250) — compile-verified
//
#include <hip/hip_runtime.h>
#include <math.h>

// ---------------------------------------------------------------------------
// MenuLoss on gfx1250.
// Dominant math: acc[k] = sum_n coeffs[k][n] * T_n(x)  == skinny f32 GEMM
// [5x447] x [447 x 172032], mapped onto V_WMMA_F32_16X16X4_F32 (full f32
// precision; low-precision WMMA would be numerically wrong for a degree-446
// Chebyshev sum). B fragments are built shuffle-free by running the T
// recurrence in both lane halves with a +2 order offset.
// ---------------------------------------------------------------------------

typedef float v2f __attribute__((ext_vector_type(2)));
typedef float v8f __attribute__((ext_vector_type(8)));

#define NCOEF 447
#define NPAD 448                 // padded to multiple of 4 (chunk K=4)
#define NCHUNK (NPAD / 4)        // 112
#define LDS_STRIDE 452           // 452 % 64 == 4 -> conflict-free b64 reads
#define BATCH 1024
#define PPB 168                  // points per batch = 7*3*8
#define PTOT (BATCH * PPB)       // 172032
#define TILES (PTOT / 16)        // 10752
#define WAVES_PER_BLOCK 8

// workspace layout (floats):
// [0 .. 5*1024)        true_menu[k][b]
// [5*1024 .. 10*1024)  pred_menu[k][b]
// [10*1024]            zp sum
// [10*1024 + 1]        idr sum
#define WS_FLOATS (10 * BATCH + 2)

__global__ __launch_bounds__(256) void menu_zero(float* ws, int n) {
  int i = blockIdx.x * 256 + threadIdx.x;
  if (i < n) ws[i] = 0.f;
}

__global__ __launch_bounds__(256) void menu_main(
    const float* __restrict__ y_pred, const float* __restrict__ y,
    const float* __restrict__ coeffs, float* __restrict__ ws) {
  __shared__ float sc[16 * LDS_STRIDE];

  // Cooperatively stage coeffs into LDS: 16 padded rows (rows 5..15 zero so
  // the unused WMMA A rows read zeros), columns 447..451 zero (K tail pad).
  for (int i = threadIdx.x; i < 16 * LDS_STRIDE; i += 256) sc[i] = 0.f;
  __syncthreads();
  for (int i = threadIdx.x; i < 5 * NCOEF; i += 256) {
    int r = i / NCOEF, c = i - r * NCOEF;
    sc[r * LDS_STRIDE + c] = coeffs[i];
  }
  __syncthreads();

  const int lane = threadIdx.x & 31;
  const int wave = threadIdx.x >> 5;
  const int tile = blockIdx.x * WAVES_PER_BLOCK + wave;
  const int half = lane >> 4;      // 0 -> K rows {0,1}; 1 -> K rows {2,3}
  const int col  = lane & 15;      // matrix column / point within tile
  const int p    = tile * 16 + col;

  // Both lane halves load the same point (coalesced within each half).
  const float pid = y_pred[2 * p + 0];
  const float pam = y_pred[2 * p + 1];
  const float tid = y[2 * p + 0];
  const float tam = y[2 * p + 1];

  // pred path: round (fwd of round_ste) then soft mask.
  const float r   = rintf(pid);
  const float sig = 1.f / (1.f + __expf(-50.f * (222.5f - r)));
  const float mid = r * sig;

  const float xt = tid * (1.f / 111.f) - 1.f;
  const float xp = mid * (1.f / 111.f) - 1.f;

  // Chebyshev state (T_{n+o}, T_{n+1+o}) with o = 2*half. Branchless init so
  // EXEC stays all-ones into the WMMA loop.
  const float t2t = 2.f * xt * xt - 1.f;
  const float t3t = 2.f * xt * t2t - xt;
  const float t2p = 2.f * xp * xp - 1.f;
  const float t3p = 2.f * xp * t2p - xp;
  float uT = half ? t2t : 1.f;
  float vT = half ? t3t : xt;
  float uP = half ? t2p : 1.f;
  float vP = half ? t3p : xp;

  const int abase = col * LDS_STRIDE + 2 * half;  // bytes*4 -> 8B aligned

  v8f cT = {};
  v8f cP = {};
  for (int ch = 0; ch < NCHUNK; ++ch) {
    v2f a = *(const v2f*)&sc[abase + 4 * ch];   // ds_load_b64, bank-clean
    v2f bT; bT.x = uT; bT.y = vT;
    v2f bP; bP.x = uP; bP.y = vP;
    cT = __builtin_amdgcn_wmma_f32_16x16x4_f32(false, a, false, bT,
                                               (short)0, cT, false, false);
    cP = __builtin_amdgcn_wmma_f32_16x16x4_f32(false, a, false, bP,
                                               (short)0, cP, false, false);
#pragma unroll
    for (int s = 0; s < 4; ++s) {               // advance 4 orders, VALU
      float wT = 2.f * xt * vT - uT; uT = vT; vT = wT;
      float wP = 2.f * xp * vP - uP; uP = vP; vP = wP;
    }
  }

  // Elementwise penalties (forward values).
  float zp  = 2500.f * (1.f - tanhf(4.f * pid)) * 2.f +
              2500.f * (1.f - tanhf(4.f * pam));
  float idr = fmaxf(pid - 222.f, 0.f);

  if (half == 0) {
    // Lane L (<16) holds C rows M=0..7 in cX[0..7]; nutrients are M=0..4.
    const int b = p / PPB;
    const float wtT = tam * (1.f / 700.f);   // /100 then /7
    const float wtP = pam * (1.f / 700.f);
#pragma unroll
    for (int k = 0; k < 5; ++k) {
      atomicAdd(&ws[k * BATCH + b],             cT[k] * wtT);
      atomicAdd(&ws[5 * BATCH + k * BATCH + b], cP[k] * wtP);
    }
    // Reduce zp/idr across the 16 owning lanes, then one atomic each.
#pragma unroll
    for (int off = 8; off > 0; off >>= 1) {
      zp  += __shfl_down(zp,  (unsigned)off, 32);
      idr += __shfl_down(idr, (unsigned)off, 32);
    }
    if (col == 0) {
      atomicAdd(&ws[10 * BATCH],     zp);
      atomicAdd(&ws[10 * BATCH + 1], idr);
    }
  }
}

__global__ __launch_bounds__(256) void menu_final(const float* __restrict__ ws,
                                                  float* __restrict__ out) {
  __shared__ float red[256];
  float acc = 0.f;
  for (int b = threadIdx.x; b < BATCH; b += 256) {
    const float scale2[5] = {1e-4f, 1e-4f, 1.f, 1.f, 1.f};
    float s = 0.f;
#pragma unroll
    for (int k = 0; k < 5; ++k) {
      float d = ws[k * BATCH + b] - ws[5 * BATCH + k * BATCH + b];
      s += scale2[k] * d * d;
    }
    acc += s;
  }
  red[threadIdx.x] = acc;
  __syncthreads();
  for (int s = 128; s > 0; s >>= 1) {
    if (threadIdx.x < s) red[threadIdx.x] += red[threadIdx.x + s];
    __syncthreads();
  }
  if (threadIdx.x == 0) {
    float inv_b = 1.f / (float)BATCH;
    float nut = red[0] * inv_b;                      // mean over batch
    out[0] = nut + ws[10 * BATCH] * inv_b + ws[10 * BATCH + 1] * inv_b;
  }
}

extern "C" void kernel_launch(void* const* d_in, const int* in_sizes, int n_in,
                              void* d_out, int out_size, void* d_ws,
                              size_t ws_size, hipStream_t stream) {
  (void)in_sizes; (void)n_in; (void)out_size; (void)ws_size;
  const float* y_pred = (const float*)d_in[0];
  const float* y      = (const float*)d_in[1];
  const float* coeffs = (const float*)d_in[2];
  float* ws  = (float*)d_ws;
  float* out = (float*)d_out;

  menu_zero<<<(WS_FLOATS + 255) / 256, 256, 0, stream>>>(ws, WS_FLOATS);
  menu_main<<<TILES / WAVES_PER_BLOCK, 256, 0, stream>>>(y_pred, y, coeffs, ws);
  menu_final<<<1, 256, 0, stream>>>(ws, out);
}